// Recursive_56418690400654
// MI455X (gfx1250) — compile-verified
//
#include <hip/hip_runtime.h>
#include <hip/hip_bf16.h>
#include <math.h>

typedef __attribute__((ext_vector_type(2))) float v2f;
typedef __attribute__((ext_vector_type(8))) float v8f;

#define LSEQ   512          // sequence length
#define BATCH  256          // batch columns
#define HID    512          // hidden size
#define COLS   16           // batch columns per workgroup (= WMMA M)
#define HMAX   6            // on-chip stack height (grammar depth <= 3)
#define BLOCK  512          // 16 wave32s
#define NT     2            // N-tiles (16 wide) per wave: 32 tiles / 16 waves

__global__ __launch_bounds__(BLOCK, 1)
void Recursive_stack_wmma_kernel(const int* __restrict__ tokens,
                                 const float* __restrict__ emb,
                                 const float* __restrict__ W,
                                 const float* __restrict__ bias,
                                 float* __restrict__ out)
{
    extern __shared__ float stackF[];           // [COLS][HMAX][HID] fp32
    __shared__ int tokLds[COLS];
    __shared__ int ptrLds[COLS];

    const int tid   = threadIdx.x;
    const int b0    = blockIdx.x * COLS;
    const int lane  = tid & 31;
    const int wv    = tid >> 5;                 // wave id 0..15
    const int khalf = lane >> 4;                // 0: lanes 0-15, 1: lanes 16-31
    const int nlane = lane & 15;                // N within tile / A-row (batch col)

    // zero-init stack (reference stack0 = zeros) and pointers
    for (int i = tid; i < COLS * HMAX * HID; i += BLOCK) stackF[i] = 0.0f;
    if (tid < COLS) ptrLds[tid] = 0;
    __syncthreads();

    // per-lane N indices + bias for the two N-tiles this wave owns
    const int   n0 = (wv +  0) * 16 + nlane;
    const int   n1 = (wv + 16) * 16 + nlane;
    const float bias0 = bias[n0];
    const float bias1 = bias[n1];
    const float* __restrict__ W0 = W + n0;
    const float* __restrict__ W1 = W + n1;

    for (int t = 0; t < LSEQ; ++t) {
        if (tid < COLS) tokLds[tid] = tokens[t * BATCH + b0 + tid];
        __syncthreads();

        // block-uniform classification masks
        unsigned pushMask = 0, closeMask = 0;
#pragma unroll
        for (int c = 0; c < COLS; ++c) {
            int tk = tokLds[c];
            if (tk != 0 && tk != 1 && tk != 2) pushMask  |= (1u << c); // real token
            if (tk == 1)                        closeMask |= (1u << c); // CLOSE
        }

        // ---- PUSH: gather embedding rows into stack[c][ptr] (float4) ----
        if (pushMask) {
            for (int idx = tid; idx < COLS * (HID / 4); idx += BLOCK) {
                int c = idx >> 7;               // HID/4 == 128
                int q = idx & 127;
                if ((pushMask >> c) & 1u) {
                    int tk = tokLds[c];
                    float4 v = ((const float4*)(emb + (size_t)tk * HID))[q];
                    ((float4*)(stackF + (size_t)(c * HMAX + ptrLds[c]) * HID))[q] = v;
                }
            }
        }

        // ---- CLOSE: parent = tanh([l;r] @ W + b) via V_WMMA_F32_16X16X4_F32 ----
        // Both N-tiles share one A-fragment load per K-step; two independent
        // accumulator chains keep the matrix pipe busy.
        if (closeMask) {
            const int arow = nlane;             // A-matrix row = batch column
            int pRow = ptrLds[arow];
            int pl = pRow - 2; if (pl < 0) pl = 0;
            int pr = pRow - 1; if (pr < 0) pr = 0;
            const float* lrow = stackF + (size_t)(arow * HMAX + pl) * HID;
            const float* rrow = stackF + (size_t)(arow * HMAX + pr) * HID;

            v8f acc0 = {0.f, 0.f, 0.f, 0.f, 0.f, 0.f, 0.f, 0.f};
            v8f acc1 = {0.f, 0.f, 0.f, 0.f, 0.f, 0.f, 0.f, 0.f};

            // left child: K = 0..511 of [l;r]
#pragma unroll 4
            for (int k0 = 0; k0 < HID; k0 += 4) {
                int kk = k0 + khalf * 2;
                v2f a = *(const v2f*)(lrow + kk);
                v2f b0v, b1v;
                b0v.x = W0[(size_t)kk * HID];
                b0v.y = W0[(size_t)(kk + 1) * HID];
                b1v.x = W1[(size_t)kk * HID];
                b1v.y = W1[(size_t)(kk + 1) * HID];
                acc0 = __builtin_amdgcn_wmma_f32_16x16x4_f32(
                    false, a, false, b0v, (short)0, acc0, false, false);
                acc1 = __builtin_amdgcn_wmma_f32_16x16x4_f32(
                    false, a, false, b1v, (short)0, acc1, false, false);
            }
            // right child: K = 512..1023 of [l;r]
#pragma unroll 4
            for (int k0 = 0; k0 < HID; k0 += 4) {
                int kk = k0 + khalf * 2;
                v2f a = *(const v2f*)(rrow + kk);
                v2f b0v, b1v;
                b0v.x = W0[(size_t)(HID + kk) * HID];
                b0v.y = W0[(size_t)(HID + kk + 1) * HID];
                b1v.x = W1[(size_t)(HID + kk) * HID];
                b1v.y = W1[(size_t)(HID + kk + 1) * HID];
                acc0 = __builtin_amdgcn_wmma_f32_16x16x4_f32(
                    false, a, false, b0v, (short)0, acc0, false, false);
                acc1 = __builtin_amdgcn_wmma_f32_16x16x4_f32(
                    false, a, false, b1v, (short)0, acc1, false, false);
            }

            __syncthreads();   // all waves done reading stack (A) before writes

            // C layout: lanes 0-15 -> M = v, lanes 16-31 -> M = v + 8; N = n0/n1
#pragma unroll
            for (int v = 0; v < 8; ++v) {
                int m = v + khalf * 8;          // batch column in tile
                if ((closeMask >> m) & 1u) {
                    int pd = ptrLds[m] - 2; if (pd < 0) pd = 0;
                    float* dst = stackF + (size_t)(m * HMAX + pd) * HID;
                    dst[n0] = tanhf(acc0[v] + bias0);
                    dst[n1] = tanhf(acc1[v] + bias1);
                }
            }
        }
        __syncthreads();

        // pointer update (push/close mutually exclusive per column)
        if (tid < COLS) {
            int p = ptrLds[tid];
            if ((pushMask >> tid) & 1u)       p += 1;
            else if ((closeMask >> tid) & 1u) p -= 1;
            ptrLds[tid] = p;
        }
        __syncthreads();
    }

    // emit stack[:, 0] -> out[b, :]
    for (int idx = tid; idx < COLS * (HID / 4); idx += BLOCK) {
        int c = idx >> 7;
        int q = idx & 127;
        float4 v = ((const float4*)(stackF + (size_t)(c * HMAX) * HID))[q];
        ((float4*)(out + (size_t)(b0 + c) * HID))[q] = v;
    }
}

extern "C" void kernel_launch(void* const* d_in, const int* in_sizes, int n_in,
                              void* d_out, int out_size, void* d_ws, size_t ws_size,
                              hipStream_t stream) {
    (void)in_sizes; (void)n_in; (void)out_size; (void)d_ws; (void)ws_size;
    const int*   tokens = (const int*)d_in[0];   // [L, B]
    const float* emb    = (const float*)d_in[1]; // [VOCAB, H]
    const float* W      = (const float*)d_in[2]; // [2H, H]
    const float* bias   = (const float*)d_in[3]; // [H]
    float* out = (float*)d_out;                  // [B, H]

    size_t smem = (size_t)COLS * HMAX * HID * sizeof(float); // 192 KB < 320 KB WGP LDS
    Recursive_stack_wmma_kernel<<<BATCH / COLS, BLOCK, smem, stream>>>(
        tokens, emb, W, bias, out);
}